// StaticProxyNCA_41429254537746
// MI455X (gfx1250) — compile-verified
//
#include <hip/hip_runtime.h>
#include <math.h>

// Problem constants (match reference)
#define N_S   4096
#define C_P   512
#define ZD    512
#define EPSW  1e-6f
#define NA    1365           // len(range(0, N-3, 3))
#define NBLK  86             // ceil(NA/16)
#define PSTR  516            // LDS stride for 16x512 A-panel (floats), bank-conflict free

typedef float v2f __attribute__((ext_vector_type(2)));
typedef float v8f __attribute__((ext_vector_type(8)));

__device__ __forceinline__ v8f wmma4(v2f a, v2f b, v8f c) {
  // V_WMMA_F32_16X16X4_F32 : D[16x16] = A[16x4] * B[4x16] + C
  return __builtin_amdgcn_wmma_f32_16x16x4_f32(false, a, false, b, (short)0, c, false, false);
}

// ---------------- Stage 1: normalize proxies + per-row (n2, sum) of normalized rows
__global__ __launch_bounds__(256) void k_prox(const float* __restrict__ pr,
                                              float* __restrict__ prox,
                                              float* __restrict__ pn2,
                                              float* __restrict__ psum) {
  const int c = blockIdx.x, t = threadIdx.x;
  const float* row = pr + (size_t)c * ZD;
  float x0 = row[t], x1 = row[t + 256];
  float ssq = x0 * x0 + x1 * x1, s = x0 + x1;
  #pragma unroll
  for (int m = 1; m < 32; m <<= 1) { ssq += __shfl_xor(ssq, m, 32); s += __shfl_xor(s, m, 32); }
  __shared__ float lq[8], ls[8];
  const int w = t >> 5, ln = t & 31;
  if (ln == 0) { lq[w] = ssq; ls[w] = s; }
  __syncthreads();
  if (t == 0) {
    float a = 0.f, b = 0.f;
    for (int i = 0; i < 8; i++) { a += lq[i]; b += ls[i]; }
    lq[0] = a; ls[0] = b;
  }
  __syncthreads();
  ssq = lq[0]; s = ls[0];
  const float rinv = 1.0f / fmaxf(sqrtf(ssq), 1e-12f);
  prox[(size_t)c * ZD + t]       = x0 * rinv;
  prox[(size_t)c * ZD + t + 256] = x1 * rinv;
  if (t == 0) { pn2[c] = ssq * rinv * rinv; psum[c] = s * rinv; }
}

// ---------------- Stage 2: per-sample (||z||^2, sum z)
__global__ __launch_bounds__(256) void k_zstat(const float* __restrict__ z,
                                               float* __restrict__ zn2,
                                               float* __restrict__ zsum) {
  const int j = blockIdx.x, t = threadIdx.x;
  const float* row = z + (size_t)j * ZD;
  float x0 = row[t], x1 = row[t + 256];
  float ssq = x0 * x0 + x1 * x1, s = x0 + x1;
  #pragma unroll
  for (int m = 1; m < 32; m <<= 1) { ssq += __shfl_xor(ssq, m, 32); s += __shfl_xor(s, m, 32); }
  __shared__ float lq[8], ls[8];
  const int w = t >> 5, ln = t & 31;
  if (ln == 0) { lq[w] = ssq; ls[w] = s; }
  __syncthreads();
  if (t == 0) {
    float a = 0.f, b = 0.f;
    for (int i = 0; i < 8; i++) { a += lq[i]; b += ls[i]; }
    zn2[j] = a; zsum[j] = b;
  }
}

// ---------------- Stage 3: label mapping
__global__ void k_inv(const int* __restrict__ y2r, int* __restrict__ inv) {
  const int i = blockIdx.x * 256 + threadIdx.x;
  if (i < C_P) inv[y2r[i]] = i;
}
__global__ void k_yrel(const int* __restrict__ yidx, const int* __restrict__ inv,
                       int* __restrict__ yrel) {
  const int j = blockIdx.x * 256 + threadIdx.x;
  if (j < N_S) yrel[j] = inv[yidx[j]];
}

// ---- epilogue helper for GEMM1: distance + validity mask + half-wave argmax
__device__ __forceinline__ void epi_argmax(v8f acc, int n, int abase, int hi,
                                           const float* __restrict__ zn2,
                                           const float* __restrict__ zsum,
                                           const int* __restrict__ yrel,
                                           const float* astat_s, const int* ya_s,
                                           float ZE2, float bv[8], int bj[8]) {
  const float zb = zn2[n] - 2.0f * EPSW * zsum[n];
  const int yn = yrel[n];
  #pragma unroll
  for (int r = 0; r < 8; r++) {
    const int m = r + 8 * hi;
    float d2 = -2.0f * acc[r] + astat_s[m] + zb + ZE2;
    float d = sqrtf(fmaxf(d2, 0.0f));
    bool valid = (n >= 3 * (abase + m)) && (yn == ya_s[m]);
    float v = valid ? d : -INFINITY;
    int j = n;
    #pragma unroll
    for (int msk = 1; msk <= 8; msk <<= 1) {   // reduce within 16-lane half
      float ov = __shfl_xor(v, msk, 32);
      int   oj = __shfl_xor(j, msk, 32);
      if (ov > v || (ov == v && oj < j)) { v = ov; j = oj; }
    }
    if (v > bv[r] || (v == bv[r] && j < bj[r])) { bv[r] = v; bj[r] = j; }
  }
}

// ---------------- Stage 4: fused pdist(P, z) + masked argmax  (WMMA GEMM1)
// 16 anchors per block; each wave register-blocks 4 N-tiles (16x64) so one
// LDS A-fragment feeds 4 WMMAs; B fragments come straight from L2-resident z.
__global__ __launch_bounds__(256) void k_d1(const float* __restrict__ z,
                                            const float* __restrict__ prox,
                                            const float* __restrict__ pn2,
                                            const float* __restrict__ psum,
                                            const float* __restrict__ zn2,
                                            const float* __restrict__ zsum,
                                            const int* __restrict__ yrel,
                                            float* __restrict__ Dp,
                                            int* __restrict__ Jp) {
  __shared__ float P[16 * PSTR];
  __shared__ float astat[16];
  __shared__ int   ya[16];
  __shared__ float BV[8][16];
  __shared__ int   BJ[8][16];

  const int abase = blockIdx.x * 16;
  const int t = threadIdx.x;
  const int w = t >> 5, lane = t & 31, nl = lane & 15, hi = lane >> 4;

  if (t < 16) {
    int aidx = abase + t;
    int ac = aidx < NA ? aidx : (NA - 1);
    int y = yrel[3 * ac];
    ya[t] = y;
    astat[t] = pn2[y] + 2.0f * EPSW * psum[y];
  }
  __syncthreads();
  for (int idx = t; idx < 16 * 128; idx += 256) {   // stage gathered 16x512 A-panel
    int r = idx >> 7, c4 = idx & 127;
    float4 v = ((const float4*)(prox + (size_t)ya[r] * ZD))[c4];
    *(float4*)&P[r * PSTR + c4 * 4] = v;
  }
  __syncthreads();

  float bv[8]; int bj[8];
  #pragma unroll
  for (int r = 0; r < 8; r++) { bv[r] = -INFINITY; bj[r] = 0; }
  const float ZE2 = (float)ZD * EPSW * EPSW;
  const float* Pl = &P[nl * PSTR + 2 * hi];

  for (int ni = 0; ni < N_S / 512; ni++) {
    const int nb = ni * 512 + w * 64;
    v8f acc0 = {0.f,0.f,0.f,0.f,0.f,0.f,0.f,0.f};
    v8f acc1 = acc0, acc2 = acc0, acc3 = acc0;
    const float* b0 = z + (size_t)(nb + nl) * ZD + 2 * hi;
    const float* b1 = b0 + 16 * ZD;
    const float* b2 = b0 + 32 * ZD;
    const float* b3 = b0 + 48 * ZD;
    #pragma unroll 4
    for (int k = 0; k < ZD; k += 4) {
      v2f a = *(const v2f*)(Pl + k);
      acc0 = wmma4(a, *(const v2f*)(b0 + k), acc0);
      acc1 = wmma4(a, *(const v2f*)(b1 + k), acc1);
      acc2 = wmma4(a, *(const v2f*)(b2 + k), acc2);
      acc3 = wmma4(a, *(const v2f*)(b3 + k), acc3);
    }
    epi_argmax(acc0, nb +  0 + nl, abase, hi, zn2, zsum, yrel, astat, ya, ZE2, bv, bj);
    epi_argmax(acc1, nb + 16 + nl, abase, hi, zn2, zsum, yrel, astat, ya, ZE2, bv, bj);
    epi_argmax(acc2, nb + 32 + nl, abase, hi, zn2, zsum, yrel, astat, ya, ZE2, bv, bj);
    epi_argmax(acc3, nb + 48 + nl, abase, hi, zn2, zsum, yrel, astat, ya, ZE2, bv, bj);
  }
  if (lane == 0 || lane == 16) {
    #pragma unroll
    for (int r = 0; r < 8; r++) { BV[w][r + 8 * hi] = bv[r]; BJ[w][r + 8 * hi] = bj[r]; }
  }
  __syncthreads();
  if (t < 16 && abase + t < NA) {
    float v = BV[0][t]; int j = BJ[0][t];
    for (int ww = 1; ww < 8; ww++) {
      float ov = BV[ww][t]; int oj = BJ[ww][t];
      if (ov > v || (ov == v && oj < j)) { v = ov; j = oj; }
    }
    Dp[abase + t] = v; Jp[abase + t] = j;
  }
}

// ---- epilogue helper for GEMM2: distance -> exp(-d) accumulation
__device__ __forceinline__ void epi_expsum(v8f acc, int cg, int hi,
                                           const float* __restrict__ pn2,
                                           const float* __restrict__ psum,
                                           const float* astat_s, float ZE2,
                                           float sacc[8]) {
  const float pb = pn2[cg] - 2.0f * EPSW * psum[cg];
  #pragma unroll
  for (int r = 0; r < 8; r++) {
    float d2 = -2.0f * acc[r] + astat_s[r + 8 * hi] + pb + ZE2;
    float d = sqrtf(fmaxf(d2, 0.0f));
    sacc[r] += expf(-d);
  }
}

// ---------------- Stage 5: fused pdist(z_a, prox) + logsumexp  (WMMA GEMM2)
// 8 waves x (16x64) tiles cover all C=512 proxies in a single pass.
__global__ __launch_bounds__(256) void k_d2(const float* __restrict__ z,
                                            const float* __restrict__ prox,
                                            const float* __restrict__ pn2,
                                            const float* __restrict__ psum,
                                            const float* __restrict__ zn2,
                                            const float* __restrict__ zsum,
                                            const float* __restrict__ Dp,
                                            const int* __restrict__ Jp,
                                            float* __restrict__ losses) {
  __shared__ float Za[16 * PSTR];
  __shared__ float astat[16];
  __shared__ int   jr[16];
  __shared__ float SM[8][16];

  const int abase = blockIdx.x * 16;
  const int t = threadIdx.x;
  const int w = t >> 5, lane = t & 31, nl = lane & 15, hi = lane >> 4;

  if (t < 16) {
    int aidx = abase + t;
    int ac = aidx < NA ? aidx : (NA - 1);
    int j = Jp[ac];
    jr[t] = j;
    astat[t] = zn2[j] + 2.0f * EPSW * zsum[j];
  }
  __syncthreads();
  for (int idx = t; idx < 16 * 128; idx += 256) {   // gather hardest-positive rows
    int r = idx >> 7, c4 = idx & 127;
    float4 v = ((const float4*)(z + (size_t)jr[r] * ZD))[c4];
    *(float4*)&Za[r * PSTR + c4 * 4] = v;
  }
  __syncthreads();

  float sacc[8];
  #pragma unroll
  for (int r = 0; r < 8; r++) sacc[r] = 0.f;
  const float ZE2 = (float)ZD * EPSW * EPSW;
  const float* Al = &Za[nl * PSTR + 2 * hi];

  const int cb = w * 64;
  v8f acc0 = {0.f,0.f,0.f,0.f,0.f,0.f,0.f,0.f};
  v8f acc1 = acc0, acc2 = acc0, acc3 = acc0;
  const float* b0 = prox + (size_t)(cb + nl) * ZD + 2 * hi;
  const float* b1 = b0 + 16 * ZD;
  const float* b2 = b0 + 32 * ZD;
  const float* b3 = b0 + 48 * ZD;
  #pragma unroll 4
  for (int k = 0; k < ZD; k += 4) {
    v2f a = *(const v2f*)(Al + k);
    acc0 = wmma4(a, *(const v2f*)(b0 + k), acc0);
    acc1 = wmma4(a, *(const v2f*)(b1 + k), acc1);
    acc2 = wmma4(a, *(const v2f*)(b2 + k), acc2);
    acc3 = wmma4(a, *(const v2f*)(b3 + k), acc3);
  }
  epi_expsum(acc0, cb +  0 + nl, hi, pn2, psum, astat, ZE2, sacc);
  epi_expsum(acc1, cb + 16 + nl, hi, pn2, psum, astat, ZE2, sacc);
  epi_expsum(acc2, cb + 32 + nl, hi, pn2, psum, astat, ZE2, sacc);
  epi_expsum(acc3, cb + 48 + nl, hi, pn2, psum, astat, ZE2, sacc);

  #pragma unroll
  for (int r = 0; r < 8; r++) {
    #pragma unroll
    for (int msk = 1; msk <= 8; msk <<= 1) sacc[r] += __shfl_xor(sacc[r], msk, 32);
  }
  if (lane == 0 || lane == 16) {
    #pragma unroll
    for (int r = 0; r < 8; r++) SM[w][r + 8 * hi] = sacc[r];
  }
  __syncthreads();
  if (t < 16 && abase + t < NA) {
    float tot = 0.f;
    for (int ww = 0; ww < 8; ww++) tot += SM[ww][t];
    losses[abase + t] = Dp[abase + t] + logf(tot);   // D_p + logsumexp(-D_n)
  }
}

// ---------------- Stage 6: mean
__global__ __launch_bounds__(256) void k_mean(const float* __restrict__ losses,
                                              float* __restrict__ out) {
  const int t = threadIdx.x;
  float s = 0.f;
  for (int i = t; i < NA; i += 256) s += losses[i];
  #pragma unroll
  for (int m = 1; m < 32; m <<= 1) s += __shfl_xor(s, m, 32);
  __shared__ float ls[8];
  const int w = t >> 5, ln = t & 31;
  if (ln == 0) ls[w] = s;
  __syncthreads();
  if (t == 0) {
    float a = 0.f;
    for (int i = 0; i < 8; i++) a += ls[i];
    out[0] = a / (float)NA;
  }
}

extern "C" void kernel_launch(void* const* d_in, const int* in_sizes, int n_in,
                              void* d_out, int out_size, void* d_ws, size_t ws_size,
                              hipStream_t stream) {
  (void)in_sizes; (void)n_in; (void)out_size; (void)ws_size;
  const float* z       = (const float*)d_in[0];   // [N, Z] f32
  const float* proxies = (const float*)d_in[1];   // [C, Z] f32
  const int*   y_idx   = (const int*)d_in[2];     // [N] i32
  const int*   y_to_r  = (const int*)d_in[3];     // [C] i32
  float* out = (float*)d_out;

  // Workspace carve-up (256B aligned slabs), ~1.1 MB total
  char* ws = (char*)d_ws;
  size_t off = 0;
  auto grab = [&](size_t bytes) { char* p = ws + off; off += (bytes + 255) & ~(size_t)255; return p; };
  float* prox   = (float*)grab((size_t)C_P * ZD * 4);
  float* pn2    = (float*)grab((size_t)C_P * 4);
  float* psum   = (float*)grab((size_t)C_P * 4);
  float* zn2    = (float*)grab((size_t)N_S * 4);
  float* zsum   = (float*)grab((size_t)N_S * 4);
  int*   inv    = (int*)  grab((size_t)C_P * 4);
  int*   yrel   = (int*)  grab((size_t)N_S * 4);
  float* Dp     = (float*)grab((size_t)NBLK * 16 * 4);
  int*   Jp     = (int*)  grab((size_t)NBLK * 16 * 4);
  float* losses = (float*)grab((size_t)NBLK * 16 * 4);

  k_prox <<<C_P, 256, 0, stream>>>(proxies, prox, pn2, psum);
  k_zstat<<<N_S, 256, 0, stream>>>(z, zn2, zsum);
  k_inv  <<<(C_P + 255) / 256, 256, 0, stream>>>(y_to_r, inv);
  k_yrel <<<(N_S + 255) / 256, 256, 0, stream>>>(y_idx, inv, yrel);
  k_d1   <<<NBLK, 256, 0, stream>>>(z, prox, pn2, psum, zn2, zsum, yrel, Dp, Jp);
  k_d2   <<<NBLK, 256, 0, stream>>>(z, prox, pn2, psum, zn2, zsum, Dp, Jp, losses);
  k_mean <<<1, 256, 0, stream>>>(losses, out);
}